// RPNProposal_53145925320991
// MI455X (gfx1250) — compile-verified
//
#include <hip/hip_runtime.h>
#include <hip/hip_bf16.h>

// ---------------- problem constants (match reference) ----------------
#define BATCH     8
#define HH        64
#define WW        100
#define NA        9
#define NANCH     (HH * WW * NA)       // 57600 anchors per image
#define PRE_NMS   2048
#define POST_NMS  300
#define NMS_T     0.7f
#define STRIDE_F  16.0f
#define HW        (HH * WW)

typedef float v2f __attribute__((ext_vector_type(2)));
typedef float v8f __attribute__((ext_vector_type(8)));

// =====================================================================
// Kernel 1: anchors + bbox decode + clip + fg-score extraction.
// Pure streaming pass: reads deltas/scores once, writes proposals+scores
// to workspace and the anchors output (batch 0 only).
// =====================================================================
__global__ void rpn_prep_kernel(const float* __restrict__ scores,
                                const float* __restrict__ deltas,
                                const float* __restrict__ im_info,
                                const float* __restrict__ bases,
                                float* __restrict__ anchors_out,
                                float4* __restrict__ ws_props,
                                float* __restrict__ ws_scores) {
    int idx = blockIdx.x * blockDim.x + threadIdx.x;
    if (idx >= BATCH * NANCH) return;
    int b   = idx / NANCH;
    int n   = idx - b * NANCH;
    int a   = n % NA;
    int pix = n / NA;
    int w   = pix % WW;
    int h   = pix / WW;

    float sx = (float)w * STRIDE_F;
    float sy = (float)h * STRIDE_F;
    float x1 = bases[a * 4 + 0] + sx;
    float y1 = bases[a * 4 + 1] + sy;
    float x2 = bases[a * 4 + 2] + sx;
    float y2 = bases[a * 4 + 3] + sy;

    if (b == 0) {
        float4* ao = (float4*)(anchors_out + (size_t)n * 4);
        *ao = make_float4(x1, y1, x2, y2);
    }

    // deltas layout (B, 36, H, W); channel c = a*4 + k
    size_t dbase = ((size_t)b * (4 * NA) + (size_t)a * 4) * HW + (size_t)h * WW + w;
    float dx = deltas[dbase];
    float dy = deltas[dbase + 1 * HW];
    float dw = deltas[dbase + 2 * HW];
    float dh = deltas[dbase + 3 * HW];

    float width  = x2 - x1 + 1.0f;
    float height = y2 - y1 + 1.0f;
    float cx = x1 + 0.5f * width;
    float cy = y1 + 0.5f * height;
    float pcx = dx * width + cx;
    float pcy = dy * height + cy;
    float pw = expf(dw) * width;
    float ph = expf(dh) * height;

    float bx1 = pcx - 0.5f * pw;
    float by1 = pcy - 0.5f * ph;
    float bx2 = pcx + 0.5f * pw;
    float by2 = pcy + 0.5f * ph;

    float maxx = im_info[b * 3 + 1] - 1.0f;
    float maxy = im_info[b * 3 + 0] - 1.0f;
    bx1 = fminf(fmaxf(bx1, 0.0f), maxx);
    by1 = fminf(fmaxf(by1, 0.0f), maxy);
    bx2 = fminf(fmaxf(bx2, 0.0f), maxx);
    by2 = fminf(fmaxf(by2, 0.0f), maxy);

    ws_props[idx] = make_float4(bx1, by1, bx2, by2);   // global_store_b128

    // fg score: channel 9 + a of (B, 18, H, W)
    ws_scores[idx] = scores[((size_t)b * (2 * NA) + NA + a) * HW + (size_t)h * WW + w];
}

// =====================================================================
// Kernel 2: per-image top-k(2048) + greedy NMS + output emit.
// One workgroup (1024 threads = 32 waves) per image; everything in LDS.
// =====================================================================
__global__ void __launch_bounds__(1024)
rpn_topk_nms_kernel(const float4* __restrict__ ws_props,
                    const float* __restrict__ ws_scores,
                    float* __restrict__ out_scores,   // (B, 300, 1)
                    float* __restrict__ out_bbox) {   // (B, 300, 5)
    const int b    = blockIdx.x;
    const int tid  = threadIdx.x;
    const int lane = tid & 31;
    const int wave = tid >> 5;

    // ---- LDS (~74 KB total, WGP has 320 KB) ----
    __shared__ float s_x1[PRE_NMS], s_y1[PRE_NMS], s_x2[PRE_NMS], s_y2[PRE_NMS]; // 32 KB
    __shared__ float s_sc[PRE_NMS];                                              // 8 KB
    __shared__ unsigned long long s_buf[4096];  // phase1: sort keys; phase2: 128x64 mask (32 KB)
    __shared__ unsigned int s_keep[64];
    __shared__ unsigned int s_wordpref[65];
    __shared__ int s_cnt[2];

    unsigned int* s_hist = (unsigned int*)s_buf;        // phase 1a (overlaid, dead before keys)
    unsigned long long* s_keys = s_buf;                 // phase 1b
    unsigned int* s_mask = (unsigned int*)s_buf;        // phase 2

    const float* sc = ws_scores + (size_t)b * NANCH;

    // ---------- phase 1a: histogram threshold select ----------
    for (int i = tid; i < 2048; i += 1024) s_hist[i] = 0u;
    if (tid < 2) s_cnt[tid] = 0;
    __syncthreads();

    for (int n = tid; n < NANCH; n += 1024) {
        float s = sc[n];
        int bin = (int)(s * 2048.0f);
        bin = bin > 2047 ? 2047 : (bin < 0 ? 0 : bin);
        atomicAdd(&s_hist[bin], 1u);
    }
    __syncthreads();

    if (tid == 0) {
        unsigned int cum = 0; int T = 0;
        for (int i = 2047; i >= 0; --i) {
            cum += s_hist[i];
            if (cum >= PRE_NMS) { T = i; break; }
        }
        s_cnt[1] = T;
    }
    __syncthreads();
    const int T = s_cnt[1];
    __syncthreads();   // hist dead; s_buf now owned by keys

    // ---------- phase 1b: compact candidates ----------
    for (int n = tid; n < NANCH; n += 1024) {
        float s = sc[n];
        int bin = (int)(s * 2048.0f);
        bin = bin > 2047 ? 2047 : (bin < 0 ? 0 : bin);
        if (bin >= T) {
            int pos = atomicAdd(&s_cnt[0], 1);
            if (pos < 4096) {
                unsigned int bits = __float_as_uint(s);
                s_keys[pos] = ((unsigned long long)bits << 32) |
                              (unsigned long long)(~(unsigned int)n);
            }
        }
    }
    __syncthreads();
    int cnt = s_cnt[0]; cnt = cnt > 4096 ? 4096 : cnt;
    for (int i = tid; i < 4096; i += 1024)
        if (i >= cnt) s_keys[i] = 0ull;

    // ---------- phase 1c: bitonic sort 4096 keys, descending ----------
    for (unsigned int k = 2; k <= 4096; k <<= 1) {
        for (unsigned int j = k >> 1; j > 0; j >>= 1) {
            __syncthreads();
            for (unsigned int i = tid; i < 4096; i += 1024) {
                unsigned int ixj = i ^ j;
                if (ixj > i) {
                    unsigned long long a = s_keys[i], c = s_keys[ixj];
                    bool desc = ((i & k) == 0);
                    if (desc ? (a < c) : (a > c)) { s_keys[i] = c; s_keys[ixj] = a; }
                }
            }
        }
    }
    __syncthreads();

    // ---------- phase 1d: gather top-2048 boxes/scores into LDS ----------
    for (int i = tid; i < PRE_NMS; i += 1024) {
        unsigned long long key = s_keys[i];
        unsigned int n = ~(unsigned int)(key & 0xFFFFFFFFull);
        float4 bx = ws_props[(size_t)b * NANCH + n];    // global_load_b128 gather
        s_x1[i] = bx.x; s_y1[i] = bx.y; s_x2[i] = bx.z; s_y2[i] = bx.w;
        s_sc[i] = __uint_as_float((unsigned int)(key >> 32));
    }
    if (tid < 64) s_keep[tid] = 0xFFFFFFFFu;
    __syncthreads();   // s_buf now owned by suppression mask

    // ---------- phase 2: NMS ----------
    // ovr > t  <=>  inter*(1+t) > t*(area_i + area_j)
    // The RHS 16x16 tile is a rank-2 outer product -> V_WMMA_F32_16X16X4_F32.
    const float t  = NMS_T;
    const float t1 = 1.0f + NMS_T;

    for (int blk = 0; blk < PRE_NMS / 128; ++blk) {      // 16 row-blocks of 128
        const int r0 = blk * 128;
        for (int i = tid; i < 128 * 64; i += 1024) s_mask[i] = 0u;
        __syncthreads();

        // Triangle skip: only columns j > r0 can ever suppress rows i >= r0,
        // so col tiles tc < r0/16 are dead (their scan 'allowed' mask is 0).
        const int tc0   = r0 >> 4;            // first useful col tile
        const int ntc   = 128 - tc0;          // useful col tiles
        const int ntile = 8 * ntc;            // 8 row-tiles x ntc col-tiles

        for (int tile = wave; tile < ntile; tile += 32) {   // wave-uniform
            const int tr = tile / ntc;                      // row tile (0..7)
            const int tc = tc0 + (tile - tr * ntc);         // col tile (tc0..127)
            const int l    = lane & 15;
            const int half = lane >> 4;
            const int m_base = r0 + tr * 16;
            const int n_base = tc * 16;

            // branchless zeroing of lanes 16..31 (they carry K=2,3 of A / B)
            const float hs = (half == 0) ? 1.0f : 0.0f;
            const int mi = m_base + l;
            const int ni = n_base + l;
            const float area_m = (s_x2[mi] - s_x1[mi] + 1.0f) * (s_y2[mi] - s_y1[mi] + 1.0f);
            const float area_n = (s_x2[ni] - s_x1[ni] + 1.0f) * (s_y2[ni] - s_y1[ni] + 1.0f);

            // A (16x4): col0 = area_i, col1 = 1      B (4x16): row0 = t, row1 = t*area_j
            v2f A; A.x = area_m * hs;  A.y = hs;
            v2f Bv; Bv.x = t * hs;     Bv.y = t * area_n * hs;
            v8f C = {};
            v8f D;
#if __has_builtin(__builtin_amdgcn_wmma_f32_16x16x4_f32)
            D = __builtin_amdgcn_wmma_f32_16x16x4_f32(false, A, false, Bv,
                                                      (short)0, C, false, false);
#else
            // VALU fallback: D[r] = t*(area[m_base+r+8*half] + area[n_base+l])
            #pragma unroll
            for (int r = 0; r < 8; ++r) {
                int m = m_base + r + 8 * half;
                float am = (s_x2[m] - s_x1[m] + 1.0f) * (s_y2[m] - s_y1[m] + 1.0f);
                D[r] = t * (am + area_n);
            }
#endif
            // this lane's column for the intersection test
            const float x1c = s_x1[ni], y1c = s_y1[ni], x2c = s_x2[ni], y2c = s_y2[ni];
            const int  wword = tc >> 1;
            const int  shift = (tc & 1) * 16;

            #pragma unroll
            for (int r = 0; r < 8; ++r) {
                const int m = m_base + r + 8 * half;   // WMMA C-layout row for D[r]
                float xx1 = fmaxf(s_x1[m], x1c);
                float yy1 = fmaxf(s_y1[m], y1c);
                float xx2 = fminf(s_x2[m], x2c);
                float yy2 = fminf(s_y2[m], y2c);
                float inter = fmaxf(xx2 - xx1 + 1.0f, 0.0f) *
                              fmaxf(yy2 - yy1 + 1.0f, 0.0f);
                bool sup = inter * t1 > D[r];
                unsigned int ball = __builtin_amdgcn_ballot_w32(sup);
                if (lane == 0) {
                    // low 16 bits: row (tr*16+r); high 16: row (tr*16+r+8)
                    int rl0 = tr * 16 + r;
                    int rl1 = rl0 + 8;
                    atomicOr(&s_mask[rl0 * 64 + wword], (ball & 0xFFFFu) << shift);
                    atomicOr(&s_mask[rl1 * 64 + wword], (ball >> 16) << shift);
                }
            }
        }
        __syncthreads();

        // serial greedy scan over this row-block: one wave, 2 keep-words/lane
        if (wave == 0) {
            volatile unsigned int* vkeep = s_keep;
            for (int rl = 0; rl < 128; ++rl) {
                const int i = r0 + rl;
                unsigned int kw = vkeep[i >> 5];           // uniform read
                if ((kw >> (i & 31)) & 1u) {
                    const int Wd = i >> 5;
                    #pragma unroll
                    for (int q = 0; q < 2; ++q) {
                        const int w = lane * 2 + q;
                        unsigned int mrow = s_mask[rl * 64 + w];
                        unsigned int allowed;
                        if (w < Wd) allowed = 0u;
                        else if (w == Wd)
                            allowed = ((i & 31) == 31) ? 0u
                                     : (0xFFFFFFFFu << ((i & 31) + 1));
                        else allowed = 0xFFFFFFFFu;
                        vkeep[w] &= ~(mrow & allowed);
                    }
                }
            }
        }
        __syncthreads();
    }

    // ---------- phase 3: ranks + emit ----------
    if (tid == 0) {
        unsigned int total = 0;
        for (int w = 0; w < 64; ++w) { s_wordpref[w] = total; total += __popc(s_keep[w]); }
        s_wordpref[64] = total;
    }
    __syncthreads();

    const int num = (int)s_wordpref[64] < POST_NMS ? (int)s_wordpref[64] : POST_NMS;
    const float bidx = (float)b;

    for (int i = tid; i < PRE_NMS; i += 1024) {
        unsigned int kw = s_keep[i >> 5];
        if ((kw >> (i & 31)) & 1u) {
            int rank = (int)s_wordpref[i >> 5] +
                       __popc(kw & ((1u << (i & 31)) - 1u));
            if (rank < POST_NMS) {
                out_scores[(size_t)b * POST_NMS + rank] = s_sc[i];
                float* bo = out_bbox + ((size_t)b * POST_NMS + rank) * 5;
                bo[1] = s_x1[i]; bo[2] = s_y1[i]; bo[3] = s_x2[i]; bo[4] = s_y2[i];
            }
        }
    }
    for (int r = tid; r < POST_NMS; r += 1024) {
        float* bo = out_bbox + ((size_t)b * POST_NMS + r) * 5;
        bo[0] = bidx;                                   // batch index column, all rows
        if (r >= num) {
            out_scores[(size_t)b * POST_NMS + r] = bidx;
            bo[1] = 0.0f; bo[2] = 0.0f; bo[3] = 0.0f; bo[4] = 0.0f;
        }
    }
}

// =====================================================================
extern "C" void kernel_launch(void* const* d_in, const int* in_sizes, int n_in,
                              void* d_out, int out_size, void* d_ws, size_t ws_size,
                              hipStream_t stream) {
    const float* scores  = (const float*)d_in[0];  // (8, 18, 64, 100)
    const float* deltas  = (const float*)d_in[1];  // (8, 36, 64, 100)
    const float* im_info = (const float*)d_in[2];  // (8, 3)
    const float* bases   = (const float*)d_in[3];  // (9, 4)

    float* out_scores  = (float*)d_out;                       // 8*300
    float* out_bbox    = (float*)d_out + BATCH * POST_NMS;    // 8*300*5
    float* out_anchors = (float*)d_out + BATCH * POST_NMS * 6;// 57600*4

    float4* ws_props = (float4*)d_ws;                                 // 8*57600 float4
    float* ws_scores = (float*)d_ws + (size_t)BATCH * NANCH * 4;      // 8*57600 f32
    (void)in_sizes; (void)n_in; (void)out_size; (void)ws_size;

    const int total = BATCH * NANCH;
    rpn_prep_kernel<<<(total + 255) / 256, 256, 0, stream>>>(
        scores, deltas, im_info, bases, out_anchors, ws_props, ws_scores);

    rpn_topk_nms_kernel<<<BATCH, 1024, 0, stream>>>(
        ws_props, ws_scores, out_scores, out_bbox);
}